// DistanceLoss_3058016715400
// MI455X (gfx1250) — compile-verified
//
#include <hip/hip_runtime.h>

typedef __attribute__((ext_vector_type(2))) float v2f;
typedef __attribute__((ext_vector_type(8))) float v8f;

#define NPTS 16384
#define TILE 16
#define ROW_TILES (NPTS / TILE)          // 1024
#define COL_TILES (NPTS / TILE)          // 1024
#define WAVES_PER_BLOCK 8
#define BLOCK_THREADS 256

// ---------------------------------------------------------------------------
// Kernel 1: rigid transform + build
//   ppack[i] = {p.x, p.y, p.z, |p|^2}            (A-side + row norms)
//   bbuf: B-fragment-ready layout, 32 v2f per column tile:
//     bbuf[ct*32 + c]      = {g.x, g.y}   (K0,K1 for lanes 0-15)
//     bbuf[ct*32 + 16 + c] = {g.z, |g|^2} (K2,K3 for lanes 16-31)
//   With A's K3 = 1.0, one v_wmma_f32_16x16x4_f32 yields
//   D[r][n] = |g_n|^2 - 2 p_r . g_n  using C = inline 0.
// ---------------------------------------------------------------------------
__global__ void __launch_bounds__(BLOCK_THREADS)
prep_kernel(const float* __restrict__ R, const float* __restrict__ t,
            const float* __restrict__ pm, const float* __restrict__ pg,
            float4* __restrict__ ppack, v2f* __restrict__ bbuf) {
    int i = blockIdx.x * blockDim.x + threadIdx.x;
    if (i >= NPTS) return;
    float mx = pm[3 * i + 0], my = pm[3 * i + 1], mz = pm[3 * i + 2];
    float px = R[0] * mx + R[1] * my + R[2] * mz + t[0];
    float py = R[3] * mx + R[4] * my + R[5] * mz + t[1];
    float pz = R[6] * mx + R[7] * my + R[8] * mz + t[2];
    ppack[i] = make_float4(px, py, pz, px * px + py * py + pz * pz);

    float gx = pg[3 * i + 0], gy = pg[3 * i + 1], gz = pg[3 * i + 2];
    float g2 = gx * gx + gy * gy + gz * gz;
    int ct = i >> 4;          // column tile
    int c  = i & 15;          // column within tile
    v2f lo; lo.x = gx; lo.y = gy;
    v2f hi; hi.x = gz; hi.y = g2;
    bbuf[ct * 32 + c]      = lo;
    bbuf[ct * 32 + 16 + c] = hi;
}

// ---------------------------------------------------------------------------
// Kernel 2: one wave per 16-row tile; sweep all 1024 column tiles.
// Inner loop: 1x global_load_b64 + 1x v_wmma_f32_16x16x4_f32 + 8x v_min.
// A-frag: lanes 0-15 {-2p.x, -2p.y}; lanes 16-31 {-2p.z, 1.0}.
// D layout: VGPR r -> row r (lanes 0-15) / row r+8 (lanes 16-31), col = lane%16.
// ---------------------------------------------------------------------------
__global__ void __launch_bounds__(BLOCK_THREADS)
nn_kernel(const float4* __restrict__ ppack, const v2f* __restrict__ bbuf,
          const float* __restrict__ pg, const int* __restrict__ midx,
          float* __restrict__ partial) {
    const int lane = threadIdx.x & 31;
    const int wave = threadIdx.x >> 5;
    const int rt   = blockIdx.x * WAVES_PER_BLOCK + wave;   // row-tile id
    const int row0 = rt * TILE;
    const int l15  = lane & 15;
    const bool hi  = lane >= 16;

    const float4 pp = ppack[row0 + l15];
    const bool sym = (midx[0] == 0);   // wave-uniform branch: EXEC stays all-ones

    if (sym) {
        v2f a;
        a.x = -2.0f * (hi ? pp.z : pp.x);
        a.y = hi ? 1.0f : (-2.0f * pp.y);
        const v8f czero = {0.f, 0.f, 0.f, 0.f, 0.f, 0.f, 0.f, 0.f};

        float mins[8];
#pragma unroll
        for (int r = 0; r < 8; ++r) mins[r] = 3.4e38f;

        const v2f* bl = bbuf + lane;    // lane-strided fragment stream
#pragma unroll 4
        for (int ct = 0; ct < COL_TILES; ++ct) {
            v2f b = bl[ct * 32];        // one global_load_b64 per lane
            v8f d = __builtin_amdgcn_wmma_f32_16x16x4_f32(
                false, a, false, b, (short)0, czero, false, false);
#pragma unroll
            for (int r = 0; r < 8; ++r) mins[r] = fminf(mins[r], d[r]);
        }

        // min across the 16 lanes of each half (wave32 xor shuffles)
#pragma unroll
        for (int r = 0; r < 8; ++r) {
            float v = mins[r];
            v = fminf(v, __shfl_xor(v, 1, 32));
            v = fminf(v, __shfl_xor(v, 2, 32));
            v = fminf(v, __shfl_xor(v, 4, 32));
            v = fminf(v, __shfl_xor(v, 8, 32));
            mins[r] = v;
        }

        // lane 0 holds rows row0..row0+7, lane 16 holds rows row0+8..row0+15
        float sum = 0.0f;
        if (l15 == 0) {
            const int rbase = row0 + (hi ? 8 : 0);
#pragma unroll
            for (int r = 0; r < 8; ++r) {
                float d2 = ppack[rbase + r].w + mins[r];
                sum += sqrtf(fmaxf(d2, 0.0f));
            }
        }
        sum += __shfl_xor(sum, 16, 32);
        if (lane == 0) partial[rt] = sum;
    } else {
        // non-symmetric path: direct per-point distance
        float dist = 0.0f;
        if (!hi) {
            const int row = row0 + l15;
            float dx = pp.x - pg[3 * row + 0];
            float dy = pp.y - pg[3 * row + 1];
            float dz = pp.z - pg[3 * row + 2];
            dist = sqrtf(dx * dx + dy * dy + dz * dz);
        }
        dist += __shfl_xor(dist, 1, 32);
        dist += __shfl_xor(dist, 2, 32);
        dist += __shfl_xor(dist, 4, 32);
        dist += __shfl_xor(dist, 8, 32);
        if (lane == 0) partial[rt] = dist;
    }
}

// ---------------------------------------------------------------------------
// Kernel 3: deterministic single-block reduction of 1024 partials -> mean.
// ---------------------------------------------------------------------------
__global__ void __launch_bounds__(BLOCK_THREADS)
reduce_kernel(const float* __restrict__ partial, float* __restrict__ out) {
    __shared__ float s[BLOCK_THREADS];
    float acc = 0.0f;
    for (int i = threadIdx.x; i < ROW_TILES; i += BLOCK_THREADS) acc += partial[i];
    s[threadIdx.x] = acc;
    __syncthreads();
    for (int off = BLOCK_THREADS / 2; off > 0; off >>= 1) {
        if ((int)threadIdx.x < off) s[threadIdx.x] += s[threadIdx.x + off];
        __syncthreads();
    }
    if (threadIdx.x == 0) out[0] = s[0] * (1.0f / (float)NPTS);
}

// ---------------------------------------------------------------------------
extern "C" void kernel_launch(void* const* d_in, const int* in_sizes, int n_in,
                              void* d_out, int out_size, void* d_ws, size_t ws_size,
                              hipStream_t stream) {
    const float* R    = (const float*)d_in[0];   // (1,3,3)
    const float* t    = (const float*)d_in[1];   // (1,3)
    const float* pm   = (const float*)d_in[2];   // (1,16384,3)
    const float* pg   = (const float*)d_in[3];   // (1,16384,3)
    const int*   midx = (const int*)d_in[4];     // scalar
    float* out = (float*)d_out;

    // workspace: ppack[16384] float4 | bbuf[16384*2] float (v2f) | partial[1024]
    float4* ppack   = (float4*)d_ws;
    v2f*    bbuf    = (v2f*)(ppack + NPTS);
    float*  partial = (float*)(bbuf + 2 * NPTS);

    prep_kernel<<<NPTS / BLOCK_THREADS, BLOCK_THREADS, 0, stream>>>(
        R, t, pm, pg, ppack, bbuf);
    nn_kernel<<<ROW_TILES / WAVES_PER_BLOCK, BLOCK_THREADS, 0, stream>>>(
        ppack, bbuf, pg, midx, partial);
    reduce_kernel<<<1, BLOCK_THREADS, 0, stream>>>(partial, out);
}